// LTIComplexConjAllpassFilter_22041772163080
// MI455X (gfx1250) — compile-verified
//
#include <hip/hip_runtime.h>
#include <hip/hip_bf16.h>

#define B_    32
#define T_    262144
#define NORD  16
#define NR    8
#define LCH   256            // chunk length
#define NCH   1024           // T_/LCH chunks per row
#define NCB   (NCH * B_)     // 32768 state columns (col = c*B_ + b)
#define SEG   64             // LDS staging segment (floats per chunk per stage)

// ws layout (floats):
#define WS_A     0           // a[0..16]
#define WS_B     32          // b[0..16]
#define WS_H     64          // H[LCH][16]  (4096)
#define WS_POWS  4224        // pows[k][16][16], k=0..9 : A1^(2^(8+k)) (2560)
#define WS_S0    8192        // S0[16][NCB]
#define WS_S1    (8192 + 16 * NCB)

typedef __attribute__((ext_vector_type(2)))  float        v2f;
typedef __attribute__((ext_vector_type(8)))  float        v8f;
typedef __attribute__((ext_vector_type(4)))  unsigned int v4u;
typedef __attribute__((ext_vector_type(8)))  int          v8i;
typedef __attribute__((ext_vector_type(4)))  int          v4i;

// ---------------------------------------------------------------------------
// Kernel 0: coefficients, companion-matrix powers, state-response matrix H.
// One block of 256 threads; serial parts on thread 0 (tiny).
// ---------------------------------------------------------------------------
__global__ __launch_bounds__(256) void coeff_kernel(const float* __restrict__ mlog,
                                                    const float* __restrict__ clog,
                                                    float* __restrict__ ws) {
  __shared__ float a_sh[17];
  __shared__ float A1[16][16];
  __shared__ float C1[16][16];
  __shared__ float C2[16][16];
  __shared__ float h0[17], h1[17];
  const int tid = threadIdx.x;

  if (tid == 0) {
    float acc[17], tmp[17];
    acc[0] = 1.f;
    for (int i = 1; i < 17; ++i) acc[i] = 0.f;
    int len = 1;
    for (int r = 0; r < NR; ++r) {
      float m  = 0.99f / (1.f + __expf(-mlog[r]));
      float cs = tanhf(clog[r]);
      float b1 = -2.f * m * cs;
      float b2 = m * m;
      for (int i = len + 1; i >= 0; --i) {
        float v = (i <= len - 1) ? acc[i] : 0.f;
        if (i - 1 >= 0 && i - 1 <= len - 1) v += b1 * acc[i - 1];
        if (i - 2 >= 0 && i - 2 <= len - 1) v += b2 * acc[i - 2];
        tmp[i] = v;
      }
      len += 2;
      for (int i = 0; i < len; ++i) acc[i] = tmp[i];
    }
    for (int i = 0; i < 17; ++i) {
      ws[WS_A + i] = acc[i];
      ws[WS_B + i] = acc[16 - i];   // b = reversed a
      a_sh[i] = acc[i];
    }
  }
  __syncthreads();

  // Companion matrix A1: row0 = -a[1..16], subdiagonal ones.
  {
    int r = tid >> 4, cc = tid & 15;
    float v = (r == 0) ? -a_sh[cc + 1] : ((cc == r - 1) ? 1.f : 0.f);
    A1[r][cc] = v;
    C1[r][cc] = v;
  }
  __syncthreads();

  // H[t][:] = row0(A1^(t+1)) via h'[j] = -a[j+1]*h[0] + h[j+1]
  if (tid < 16) h0[tid] = A1[0][tid];
  __syncthreads();
  {
    float* hc = h0;
    float* hn = h1;
    for (int t = 0; t < LCH; ++t) {
      if (tid < 16) {
        ws[WS_H + t * 16 + tid] = hc[tid];
        hn[tid] = -a_sh[tid + 1] * hc[0] + ((tid < 15) ? hc[tid + 1] : 0.f);
      }
      __syncthreads();
      float* tp = hc; hc = hn; hn = tp;
    }
  }

  // Repeated squaring: after j squarings C1 = A1^(2^j). Keep j = 8..17.
  {
    int r = tid >> 4, cc = tid & 15;
    for (int j = 1; j <= 17; ++j) {
      float v = 0.f;
      #pragma unroll
      for (int k = 0; k < 16; ++k) v += C1[r][k] * C1[k][cc];
      C2[r][cc] = v;
      __syncthreads();
      C1[r][cc] = C2[r][cc];
      if (j >= 8) ws[WS_POWS + (j - 8) * 256 + tid] = v;
      __syncthreads();
    }
  }
}

// ---------------------------------------------------------------------------
// Kernel 1: zero-initial-state response per chunk. One lane per (row, chunk).
// x staged into LDS with the Tensor Data Mover (strided 2D tile, LDS padding).
// Writes y0 into d_out and chunk-end states E_c into S0.
// ---------------------------------------------------------------------------
__global__ __launch_bounds__(256) void phase1_kernel(const float* __restrict__ x,
                                                     float* __restrict__ y0,
                                                     float* __restrict__ ws) {
  __shared__ float xs[256][SEG + 1];   // TDM pads 1 dword per 64 -> stride 65
  __shared__ float ys[256][SEG + 1];
  const int tid = threadIdx.x;
  const int g   = blockIdx.x * 256 + tid;
  const int b   = g >> 10;             // / NCH
  const int c   = g & (NCH - 1);
  const int b0  = (blockIdx.x * 256) >> 10;   // block-uniform (256 | NCH)
  const int c0  = (blockIdx.x * 256) & (NCH - 1);

  float ac[NORD], bcr[17];
  #pragma unroll
  for (int i = 0; i < NORD; ++i) ac[i] = ws[WS_A + 1 + i];
  #pragma unroll
  for (int i = 0; i < 17; ++i) bcr[i] = ws[WS_B + i];

  // FIR history: 16 samples before chunk start (true x, zeros before t=0).
  float xwin[NORD];
  {
    const long tstart = (long)c * LCH;
    #pragma unroll
    for (int k = 0; k < NORD; ++k) {
      long tt = tstart - 1 - k;
      xwin[k] = (tt >= 0) ? x[(size_t)b * T_ + (size_t)tt] : 0.f;
    }
  }
  float s[NORD];
  #pragma unroll
  for (int i = 0; i < NORD; ++i) s[i] = 0.f;

  const unsigned lds_base = (unsigned)(size_t)(&xs[0][0]);
  const float* gbase = x + (size_t)b0 * T_ + (size_t)c0 * LCH;

  for (int seg = 0; seg < LCH / SEG; ++seg) {
    if ((tid >> 5) == 0) {   // one wave issues the TDM load for the block
      unsigned long long ga = (unsigned long long)(size_t)(gbase + seg * SEG);
      v4u g0;
      v8i g1;
      v4i g2 = {0, 0, 0, 0};
      v4i g3 = {0, 0, 0, 0};
      v8i g4 = {0, 0, 0, 0, 0, 0, 0, 0};
      g0.x = 1u;                                        // count = 1 descriptor
      g0.y = lds_base;                                  // LDS byte address
      g0.z = (unsigned)(ga & 0xFFFFFFFFull);            // global_addr[31:0]
      g0.w = (unsigned)((ga >> 32) & 0x1FFFFFFull) | (2u << 30); // [56:32] | type=2
      const int td0 = SEG;                              // tensor_dim0
      const int td1 = 256;                              // tensor_dim1
      g1[0] = (2 << 16) | (1 << 20) | (5 << 22);        // 4B elems, pad_en, 64dw interval, 1dw pad
      g1[1] = (td0 & 0xFFFF) << 16;                     // abar=0 | tensor_dim0.lo
      g1[2] = (td0 >> 16) | ((td1 & 0xFFFF) << 16);     // tensor_dim0.hi | tensor_dim1.lo
      g1[3] = (td1 >> 16) | (SEG << 16);                // tensor_dim1.hi | tile_dim0=SEG
      g1[4] = 256;                                      // tile_dim1=256 rows, tile_dim2=0
      g1[5] = LCH;                                      // tensor_dim0_stride (elements)
      g1[6] = 0;
      g1[7] = 0;
      __builtin_amdgcn_tensor_load_to_lds(g0, g1, g2, g3, g4, 0);
      __builtin_amdgcn_s_wait_tensorcnt(0);
    }
    __syncthreads();

    #pragma unroll 4
    for (int i = 0; i < SEG; ++i) {
      float xt  = xs[tid][i];
      float fir = bcr[0] * xt;
      #pragma unroll
      for (int k = 1; k < 17; ++k) fir += bcr[k] * xwin[k - 1];
      float yv = fir;
      #pragma unroll
      for (int k = 0; k < NORD; ++k) yv -= ac[k] * s[k];
      #pragma unroll
      for (int k = NORD - 1; k > 0; --k) s[k] = s[k - 1];
      s[0] = yv;
      #pragma unroll
      for (int k = NORD - 1; k > 0; --k) xwin[k] = xwin[k - 1];
      xwin[0] = xt;
      ys[tid][i] = yv;
    }
    __syncthreads();

    // coalesced writeback of this segment
    for (int e = tid; e < 256 * SEG; e += 256) {
      int r = e >> 6, col = e & (SEG - 1);
      y0[(size_t)b0 * T_ + (size_t)(c0 + r) * LCH + seg * SEG + col] = ys[r][col];
    }
    __syncthreads();
  }

  // chunk-end state E_c = (y[L-1], ..., y[L-16])
  float* S0 = ws + WS_S0;
  #pragma unroll
  for (int j = 0; j < NORD; ++j) S0[(size_t)j * NCB + c * B_ + b] = s[j];
}

// ---------------------------------------------------------------------------
// Kernel 2: Kogge-Stone level of affine scan over chunks, via f32 WMMA.
// t'_col = P * t_(col-shift) + t_col , P = A1^(L*2^k). One 16-col tile / wave.
// ---------------------------------------------------------------------------
__global__ __launch_bounds__(256) void scan_kernel(const float* __restrict__ src,
                                                   float* __restrict__ dst,
                                                   const float* __restrict__ P,
                                                   int shiftcols) {
  const int tid  = threadIdx.x;
  const int l    = tid & 31;
  const int wave = tid >> 5;
  const int tile = blockIdx.x * 8 + wave;
  const int col0 = tile * 16;
  const int n    = l & 15;
  const int hi   = l >> 4;

  v8f acc;
  #pragma unroll
  for (int v = 0; v < 8; ++v) {
    int m = v + hi * 8;
    acc[v] = src[(size_t)m * NCB + col0 + n];
  }

  if (col0 >= shiftcols) {         // tile-uniform branch: EXEC stays all-ones
    const int scol = col0 - shiftcols + n;
    #pragma unroll
    for (int st = 0; st < 4; ++st) {
      v2f av, bv;
      av.x = P[n * 16 + 4 * st + hi * 2 + 0];                 // A[m=n][k]
      av.y = P[n * 16 + 4 * st + hi * 2 + 1];
      bv.x = src[(size_t)(4 * st + hi * 2 + 0) * NCB + scol]; // B[k][n]
      bv.y = src[(size_t)(4 * st + hi * 2 + 1) * NCB + scol];
      acc = __builtin_amdgcn_wmma_f32_16x16x4_f32(false, av, false, bv,
                                                  (short)0, acc, false, false);
    }
  }

  #pragma unroll
  for (int v = 0; v < 8; ++v) {
    int m = v + hi * 8;
    dst[(size_t)m * NCB + col0 + n] = acc[v];
  }
}

// ---------------------------------------------------------------------------
// Kernel 3: per-chunk correction y[b][t] += H[t_local] . s_in  via f32 WMMA,
// computed transposed (D rows = batch, cols = t) for coalesced y updates.
// One block per chunk c >= 1 (chunk 0 has zero initial state).
// ---------------------------------------------------------------------------
__global__ __launch_bounds__(256) void phase3_kernel(float* __restrict__ y,
                                                     const float* __restrict__ ws) {
  __shared__ float Hs[LCH][17];    // padded: conflict-free lane-varying reads
  const int c   = blockIdx.x + 1;
  const int tid = threadIdx.x;
  const float* H  = ws + WS_H;
  const float* S0 = ws + WS_S0;

  for (int e = tid; e < LCH * 16; e += 256) Hs[e >> 4][e & 15] = H[e];
  __syncthreads();

  const int l    = tid & 31;
  const int wave = tid >> 5;
  const int n    = l & 15;
  const int hi   = l >> 4;

  for (int p = wave; p < 32; p += 8) {      // 16 t-tiles x 2 b-tiles
    const int tt = p >> 1;
    const int bt = p & 1;
    const int scolbase = (c - 1) * B_ + bt * 16;   // s_in = scanned s_end(c-1)
    v8f acc = {0.f, 0.f, 0.f, 0.f, 0.f, 0.f, 0.f, 0.f};
    #pragma unroll
    for (int st = 0; st < 4; ++st) {
      v2f av, bv;
      av.x = S0[(size_t)(4 * st + hi * 2 + 0) * NCB + scolbase + n]; // A[b][k]=s_in[k][b]
      av.y = S0[(size_t)(4 * st + hi * 2 + 1) * NCB + scolbase + n];
      bv.x = Hs[tt * 16 + n][4 * st + hi * 2 + 0];                   // B[k][t]=H[t][k]
      bv.y = Hs[tt * 16 + n][4 * st + hi * 2 + 1];
      acc = __builtin_amdgcn_wmma_f32_16x16x4_f32(false, av, false, bv,
                                                  (short)0, acc, false, false);
    }
    #pragma unroll
    for (int v = 0; v < 8; ++v) {
      int bl = v + hi * 8;                                           // batch-local row
      size_t idx = (size_t)(bt * 16 + bl) * T_ + (size_t)c * LCH + tt * 16 + n;
      y[idx] += acc[v];
    }
  }
}

// ---------------------------------------------------------------------------
extern "C" void kernel_launch(void* const* d_in, const int* in_sizes, int n_in,
                              void* d_out, int out_size, void* d_ws, size_t ws_size,
                              hipStream_t stream) {
  const float* ex   = (const float*)d_in[0];
  const float* mlog = (const float*)d_in[1];
  const float* clog = (const float*)d_in[2];
  float* out = (float*)d_out;
  float* ws  = (float*)d_ws;

  coeff_kernel<<<1, 256, 0, stream>>>(mlog, clog, ws);
  phase1_kernel<<<NCB / 256, 256, 0, stream>>>(ex, out, ws);

  float* S0 = ws + WS_S0;
  float* S1 = ws + WS_S1;
  for (int k = 0; k < 10; ++k) {
    const float* P = ws + WS_POWS + k * 256;
    const float* src = (k & 1) ? S1 : S0;
    float*       dst = (k & 1) ? S0 : S1;
    scan_kernel<<<(NCB / 16) / 8, 256, 0, stream>>>(src, dst, P, 32 << k);
  }
  // after 10 levels the inclusive scan (s_end per chunk) lands back in S0

  phase3_kernel<<<NCH - 1, 256, 0, stream>>>(out, ws);
}